// WavyBert_83167746720418
// MI455X (gfx1250) — compile-verified
//
#include <hip/hip_runtime.h>
#include <stdint.h>

// Problem constants (fixed by the harness' setup_inputs)
#define Bn 64
#define Tn 512
#define Hn 768
#define Ln 400
#define HM 767   // H-1 feature channels

// ---------------- gfx1250 async global->LDS helpers (inline asm, per CDNA5 ISA) ----
__device__ __forceinline__ void async_copy_b32(unsigned lds_byte_off, const float* gaddr) {
    // GLOBAL_LOAD_ASYNC_TO_LDS_B32: vdst = LDS byte address, vaddr = 64-bit global address
    asm volatile("global_load_async_to_lds_b32 %0, %1, off"
                 :: "v"(lds_byte_off), "v"((unsigned long long)(size_t)gaddr)
                 : "memory");
}
__device__ __forceinline__ void wait_asynccnt0() {
    asm volatile("s_wait_asynccnt 0" ::: "memory");
}
__device__ __forceinline__ unsigned lds_addr_of(const void* p) {
    // flat pointer to __shared__: low 32 bits are the LDS byte offset
    return (unsigned)(unsigned long long)(size_t)p;
}

// ---------------- Kernel 1: alpha, alpha_sum, per-batch scalar fire chain ----------------
__global__ __launch_bounds__(512)
void k_alpha(const float* __restrict__ hs, const int* __restrict__ lens,
             float* __restrict__ w, float* __restrict__ c0, float* __restrict__ c1,
             float* __restrict__ nh) {
    const int b = blockIdx.x;
    const int t = threadIdx.x;               // 512 threads == T
    __shared__ float sal[Tn];
    __shared__ float red[Tn];

    float x = hs[(size_t)b * Tn * Hn + (size_t)t * Hn + (Hn - 1)];
    float a = 1.0f / (1.0f + expf(-x));      // sigmoid
    sal[t] = a;
    red[t] = a;
    __syncthreads();
    #pragma unroll
    for (int s = Tn / 2; s > 0; s >>= 1) {
        if (t < s) red[t] += red[t + s];
        __syncthreads();
    }
    if (t == 0) {
        float asum  = red[0];
        int   li    = lens[b]; if (li > Ln) li = Ln;
        float len   = (float)li;
        float scale = len / asum;
        float d     = len - asum;
        nh[b] = d * d;

        float a_r = 0.0f;
        const int base = b * Ln;
        for (int k = 0; k < Ln; ++k) {
            float ak    = sal[k] * scale;
            float aa    = ak + a_r;
            bool  fired = (aa >= 1.0f);
            float wv    = fired ? (1.0f - a_r) : ak;
            float arn   = fired ? (ak - (1.0f - a_r)) : aa;   // matches reference rounding
            w [base + k] = wv;
            c0[base + k] = fired ? 0.0f : 1.0f;
            c1[base + k] = fired ? arn  : wv;
            a_r = arn;
        }
    }
}

// ---------------- Kernel 2: n_hat reduction (deterministic, no atomics) ----------------
__global__ __launch_bounds__(64)
void k_nhat(const float* __restrict__ nh, float* __restrict__ out) {
    __shared__ float r[Bn];
    const int t = threadIdx.x;
    r[t] = nh[t];
    __syncthreads();
    #pragma unroll
    for (int s = Bn / 2; s > 0; s >>= 1) {
        if (t < s) r[t] += r[t + s];
        __syncthreads();
    }
    if (t == 0) out[(size_t)Bn * Ln * HM] = r[0];
}

// ---------------- Kernel 3: the memory-bound scan over k ----------------
// grid = (3, 64): blockIdx.y = batch, blockIdx.x * 256 + tid = feature column h.
// Coefficients (w, c0, c1)[b][0..399] staged once via async global->LDS.
// Per-thread serial dependence is only the s_r FMA chain; loads unrolled x16
// so each wave keeps 16 coalesced 128B load packets in flight.
__global__ __launch_bounds__(256)
void k_scan(const float* __restrict__ hs,
            const float* __restrict__ w, const float* __restrict__ c0,
            const float* __restrict__ c1, float* __restrict__ out) {
    const int b = blockIdx.y;
    const int h = blockIdx.x * 256 + threadIdx.x;

    __shared__ float sw[Ln];
    __shared__ float s0[Ln];
    __shared__ float s1[Ln];

    const int base = b * Ln;
    for (int i = threadIdx.x; i < Ln; i += 256) {
        async_copy_b32(lds_addr_of(&sw[i]), w  + base + i);
        async_copy_b32(lds_addr_of(&s0[i]), c0 + base + i);
        async_copy_b32(lds_addr_of(&s1[i]), c1 + base + i);
    }
    wait_asynccnt0();
    __syncthreads();

    if (h < HM) {
        const float* hp = hs  + (size_t)b * Tn * Hn + h;   // stride Hn per k (coalesced across lanes)
        float*       op = out + (size_t)b * Ln * HM + h;   // stride HM per k
        float s_r = 0.0f;
        #pragma unroll 1
        for (int k0 = 0; k0 < Ln; k0 += 16) {              // 400 = 25 * 16
            float hk[16];
            #pragma unroll
            for (int u = 0; u < 16; ++u)
                hk[u] = hp[(size_t)(k0 + u) * Hn];
            #pragma unroll
            for (int u = 0; u < 16; ++u) {
                const int k = k0 + u;
                float sa = fmaf(sw[k], hk[u], s_r);
                __builtin_nontemporal_store(sa, &op[(size_t)k * HM]);  // NT: don't evict L2-resident input
                s_r = fmaf(s1[k], hk[u], s0[k] * s_r);
            }
        }
    }
}

// ---------------- Launch ----------------
extern "C" void kernel_launch(void* const* d_in, const int* in_sizes, int n_in,
                              void* d_out, int out_size, void* d_ws, size_t ws_size,
                              hipStream_t stream) {
    (void)in_sizes; (void)n_in; (void)out_size; (void)ws_size;
    const float* hs   = (const float*)d_in[0];
    const int*   lens = (const int*)d_in[1];
    float*       out  = (float*)d_out;

    float* w  = (float*)d_ws;            // Bn*Ln floats
    float* c0 = w  + Bn * Ln;            // Bn*Ln floats
    float* c1 = c0 + Bn * Ln;            // Bn*Ln floats
    float* nh = c1 + Bn * Ln;            // Bn floats

    k_alpha<<<dim3(Bn), dim3(Tn), 0, stream>>>(hs, lens, w, c0, c1, nh);
    k_nhat <<<dim3(1),  dim3(Bn), 0, stream>>>(nh, out);
    k_scan <<<dim3(3, Bn), dim3(256), 0, stream>>>(hs, w, c0, c1, out);
}